// Attn_37005438223068
// MI455X (gfx1250) — compile-verified
//
#include <hip/hip_runtime.h>
#include <math.h>

// Problem sizes (compile-time constants from the reference)
#define SEQ 2048
#define BATCH 64
#define HID 512
#define FEAT 1024  // 2*H

typedef __attribute__((ext_vector_type(2))) float v2f;
typedef __attribute__((ext_vector_type(8))) float v8f;

// ---------------------------------------------------------------------------
// Kernel 1: v[b,f] = sum_h hidden[b,h] * W[h,f]
// hidden: [B, H] row-major, W: [H, F] row-major, v: [B, F] row-major.
// One wave32 per 16x16 output tile, K-loop in steps of 4 using
// V_WMMA_F32_16X16X4_F32.  A (16x4) layout: lanes 0-15 hold K=0,1 in
// VGPR0/1; lanes 16-31 hold K=2,3.  B (4x16) mirrors that split.
// D/C: 8 VGPRs, lane<16 -> rows M=0..7, lane>=16 -> rows M=8..15.
// ---------------------------------------------------------------------------
__global__ __launch_bounds__(32) void gemm_vw_wmma(const float* __restrict__ hidden,
                                                   const float* __restrict__ W,
                                                   float* __restrict__ v) {
    const int lane = threadIdx.x;          // 0..31
    const int tile = blockIdx.x;           // 256 tiles: 4 (M) x 64 (N)
    const int m0 = (tile >> 6) << 4;       // 0,16,32,48
    const int n0 = (tile & 63) << 4;       // 0..1008
    const int lo = lane & 15;
    const int kg = (lane >> 4) << 1;       // 0 for lanes 0-15, 2 for 16-31

    v8f acc = {};
    for (int k = 0; k < HID; k += 4) {
        v2f a, bm;
        const float* ap = hidden + (m0 + lo) * HID + (k + kg);
        a.x = ap[0];
        a.y = ap[1];
        const float* bp = W + (k + kg) * FEAT + (n0 + lo);
        bm.x = bp[0];
        bm.y = bp[FEAT];
        acc = __builtin_amdgcn_wmma_f32_16x16x4_f32(
            /*neg_a=*/false, a, /*neg_b=*/false, bm,
            /*c_mod=*/(short)0, acc, /*reuse_a=*/false, /*reuse_b=*/false);
    }

    const int rowbase = m0 + ((lane >> 4) << 3);  // +8 for upper half-wave
    float* vp = v + rowbase * FEAT + (n0 + lo);
#pragma unroll
    for (int r = 0; r < 8; ++r) vp[r * FEAT] = acc[r];
}

// ---------------------------------------------------------------------------
// Kernel 1b: hb[b] = dot(hidden[b,:], bias)   (bias folded into scores)
// ---------------------------------------------------------------------------
__global__ __launch_bounds__(32) void hbias_kernel(const float* __restrict__ hidden,
                                                   const float* __restrict__ bias,
                                                   float* __restrict__ hb) {
    const int b = blockIdx.x;
    const int lane = threadIdx.x;
    float acc = 0.f;
    for (int h = lane; h < HID; h += 32) acc += hidden[b * HID + h] * bias[h];
#pragma unroll
    for (int off = 16; off > 0; off >>= 1) acc += __shfl_down(acc, off, 32);
    if (lane == 0) hb[b] = acc;
}

// ---------------------------------------------------------------------------
// Kernel 2: scores[b,s] = dot(enc[s,b,:], v[b,:]) + hb[b]  (written to d_out)
// Bandwidth-bound: one pass over 512 MiB of encoder_outputs.
// grid = (SEQ/64, BATCH), block = 256 (8 waves); each wave owns 8 s-rows,
// lanes do coalesced float4 loads; v[b,:] staged in LDS.
// ---------------------------------------------------------------------------
__global__ __launch_bounds__(256) void score_kernel(const float4* __restrict__ enc4,
                                                    const float4* __restrict__ v4,
                                                    const float* __restrict__ hb,
                                                    float* __restrict__ out) {
    __shared__ float4 vsh[FEAT / 4];  // 256 float4 = 4 KB
    const int b = blockIdx.y;
    const int tid = threadIdx.x;

    vsh[tid] = v4[b * (FEAT / 4) + tid];
    __syncthreads();

    const int wave = tid >> 5;
    const int lane = tid & 31;
    const float hbb = hb[b];
    const int s_base = blockIdx.x * 64 + wave * 8;

    for (int r = 0; r < 8; ++r) {
        const int s = s_base + r;
        const float4* e = enc4 + ((size_t)s * BATCH + b) * (FEAT / 4);
        // Prefetch next s-row for this (b, wave) -> global_prefetch_b8
        __builtin_prefetch((const void*)(e + (size_t)BATCH * (FEAT / 4) + lane), 0, 0);
        float acc = 0.f;
#pragma unroll
        for (int k = 0; k < 8; ++k) {
            const float4 ev = e[lane + 32 * k];
            const float4 wv = vsh[lane + 32 * k];
            acc += ev.x * wv.x + ev.y * wv.y + ev.z * wv.z + ev.w * wv.w;
        }
#pragma unroll
        for (int off = 16; off > 0; off >>= 1) acc += __shfl_down(acc, off, 32);
        if (lane == 0) out[b * SEQ + s] = acc + hbb;
    }
}

// ---------------------------------------------------------------------------
// Kernel 3: softmax over s for each b, in place on d_out. 2048 elts / block.
// ---------------------------------------------------------------------------
__global__ __launch_bounds__(256) void softmax_kernel(float* __restrict__ out) {
    __shared__ float red[256];
    const int b = blockIdx.x;
    const int tid = threadIdx.x;
    float* p = out + b * SEQ;

    float x[8];
    float m = -INFINITY;
#pragma unroll
    for (int i = 0; i < 8; ++i) {
        x[i] = p[tid + (i << 8)];
        m = fmaxf(m, x[i]);
    }
    red[tid] = m;
    __syncthreads();
    for (int off = 128; off > 0; off >>= 1) {
        if (tid < off) red[tid] = fmaxf(red[tid], red[tid + off]);
        __syncthreads();
    }
    m = red[0];
    __syncthreads();

    float sum = 0.f;
#pragma unroll
    for (int i = 0; i < 8; ++i) {
        x[i] = __expf(x[i] - m);
        sum += x[i];
    }
    red[tid] = sum;
    __syncthreads();
    for (int off = 128; off > 0; off >>= 1) {
        if (tid < off) red[tid] += red[tid + off];
        __syncthreads();
    }
    const float inv = 1.0f / red[0];
#pragma unroll
    for (int i = 0; i < 8; ++i) p[tid + (i << 8)] = x[i] * inv;
}

// ---------------------------------------------------------------------------
extern "C" void kernel_launch(void* const* d_in, const int* in_sizes, int n_in,
                              void* d_out, int out_size, void* d_ws, size_t ws_size,
                              hipStream_t stream) {
    const float* hidden = (const float*)d_in[0];  // [1,B,H]
    const float* enc    = (const float*)d_in[1];  // [S,B,2H]
    const float* W      = (const float*)d_in[2];  // [H,2H]
    const float* bias   = (const float*)d_in[3];  // [H]
    float* out = (float*)d_out;                   // [B,1,S] flat = B*S

    float* v  = (float*)d_ws;          // BATCH*FEAT floats = 256 KB
    float* hb = v + BATCH * FEAT;      // BATCH floats

    // Stage 1: v = hidden @ W via WMMA (4 x 64 tiles of 16x16)
    gemm_vw_wmma<<<256, 32, 0, stream>>>(hidden, W, v);
    hbias_kernel<<<BATCH, 32, 0, stream>>>(hidden, bias, hb);

    // Stage 2: scores -> d_out (streams 512 MiB of enc at HBM rate)
    dim3 g2(SEQ / 64, BATCH);
    score_kernel<<<g2, 256, 0, stream>>>((const float4*)enc, (const float4*)v, hb, out);

    // Stage 3: softmax over seq axis, in place
    softmax_kernel<<<BATCH, 256, 0, stream>>>(out);
}